// AdditiveAttention_11063835754785
// MI455X (gfx1250) — compile-verified
//
#include <hip/hip_runtime.h>
#include <math.h>

#define B_ 2
#define S_ 1024
#define H_ 512
#define A_ 128

typedef __attribute__((ext_vector_type(2))) float v2f;
typedef __attribute__((ext_vector_type(8))) float v8f;

// Hardware tanh on gfx1250 (trans pipe). Prefer the builtin so the compiler
// handles the TRANS-op result hazard; fall back to inline asm with an
// embedded v_nop to satisfy the "1 independent op after trans" rule.
__device__ __forceinline__ float fast_tanh(float x) {
#if __has_builtin(__builtin_amdgcn_tanhf)
  return __builtin_amdgcn_tanhf(x);
#else
  float y;
  asm volatile("v_tanh_f32 %0, %1\n\tv_nop" : "=v"(y) : "v"(x));
  return y;
#endif
}

// ---------------------------------------------------------------------------
// Kernel A: WX = X*W, UX = X*U via f32 WMMA. One wave computes one 16x16
// output tile; K-loop over H in steps of 4 (v_wmma_f32_16x16x4_f32).
// A-matrix (16x4 f32) layout: lane L(0-15): vgpr0=(M=L,K=0), vgpr1=(M=L,K=1);
// lanes 16-31 hold K=2,3. B-matrix (4x16) mirrors with N striped over lanes.
// C/D: vgpr r: lanes0-15 -> (M=r, N=lane), lanes16-31 -> (M=r+8, N=lane-16).
// grid = (32, 8, 2), block = 128 (4 waves, each one M-tile).
// ---------------------------------------------------------------------------
__global__ __launch_bounds__(128) void wmma_proj_kernel(
    const float* __restrict__ X,   // (B*S, H) row-major
    const float* __restrict__ W,   // (H, A) row-major
    const float* __restrict__ U,   // (H, A) row-major
    float* __restrict__ WX,        // (B*S, A)
    float* __restrict__ UX) {      // (B*S, A)
  const int lane = threadIdx.x & 31;
  const int wave = threadIdx.x >> 5;
  const int m0 = (blockIdx.x * 4 + wave) * 16;
  const int n0 = blockIdx.y * 16;
  const float* __restrict__ Bm = blockIdx.z ? U : W;
  float* __restrict__ Out = blockIdx.z ? UX : WX;
  const int hi = lane >> 4;       // 0: K=0,1  1: K=2,3
  const int lo = lane & 15;       // M (for A) / N (for B, C/D)
  const float* __restrict__ arow = X + (size_t)(m0 + lo) * H_;

  v8f acc = {};
  for (int k0 = 0; k0 < H_; k0 += 4) {
    const int ka = k0 + hi * 2;
    v2f a, b;
    a.x = arow[ka];
    a.y = arow[ka + 1];
    b.x = Bm[(size_t)ka * A_ + n0 + lo];
    b.y = Bm[(size_t)(ka + 1) * A_ + n0 + lo];
    acc = __builtin_amdgcn_wmma_f32_16x16x4_f32(
        /*neg_a=*/false, a, /*neg_b=*/false, b,
        /*c_mod=*/(short)0, acc, /*reuse_a=*/false, /*reuse_b=*/false);
  }
#pragma unroll
  for (int r = 0; r < 8; ++r)
    Out[(size_t)(m0 + r + hi * 8) * A_ + n0 + lo] = acc[r];
}

// ---------------------------------------------------------------------------
// Kernel B: for one attention row (b,i): e[j] = sum_a v[a]*tanh(wx[a]+ux[j,a]),
// softmax over j, write normalized row. One block (256 thr = 8 waves) per row;
// thread t owns j = t, t+256, t+512, t+768. Trans-pipe bound (v_tanh_f32),
// FMA co-executes on the VALU pipe.
// ---------------------------------------------------------------------------
__global__ __launch_bounds__(256) void attn_row_kernel(
    const float* __restrict__ WX, const float* __restrict__ UX,
    const float* __restrict__ V, float* __restrict__ attn) {
  __shared__ float s_wx[A_];
  __shared__ float s_v[A_];
  __shared__ float s_red[8];

  const int row = blockIdx.x;          // b*S + i
  const int b = row >> 10;
  const int t = threadIdx.x;
  const int lane = t & 31;
  const int wid = t >> 5;

  if (t < A_) {
    s_wx[t] = WX[(size_t)row * A_ + t];
    s_v[t]  = V[t];
  }
  __syncthreads();

  const float* __restrict__ uxb = UX + (size_t)b * S_ * A_;
  float e[4];
#pragma unroll
  for (int jj = 0; jj < 4; ++jj) {
    const int j = t + jj * 256;
    const float4* __restrict__ u4 = (const float4*)(uxb + (size_t)j * A_);
    float acc = 0.f;
#pragma unroll 8
    for (int a4 = 0; a4 < A_ / 4; ++a4) {
      const float4 u = u4[a4];
      const int a = a4 * 4;
      acc = fmaf(s_v[a + 0], fast_tanh(s_wx[a + 0] + u.x), acc);
      acc = fmaf(s_v[a + 1], fast_tanh(s_wx[a + 1] + u.y), acc);
      acc = fmaf(s_v[a + 2], fast_tanh(s_wx[a + 2] + u.z), acc);
      acc = fmaf(s_v[a + 3], fast_tanh(s_wx[a + 3] + u.w), acc);
    }
    e[jj] = acc;
  }

  // block max (wave32 shuffle + LDS across 8 waves)
  float m = fmaxf(fmaxf(e[0], e[1]), fmaxf(e[2], e[3]));
#pragma unroll
  for (int off = 16; off > 0; off >>= 1)
    m = fmaxf(m, __shfl_xor(m, off, 32));
  if (lane == 0) s_red[wid] = m;
  __syncthreads();
  float bm = s_red[0];
#pragma unroll
  for (int i = 1; i < 8; ++i) bm = fmaxf(bm, s_red[i]);
  __syncthreads();

  // exp + block sum
  float p[4];
  float ls = 0.f;
#pragma unroll
  for (int jj = 0; jj < 4; ++jj) {
    p[jj] = __expf(e[jj] - bm);
    ls += p[jj];
  }
#pragma unroll
  for (int off = 16; off > 0; off >>= 1)
    ls += __shfl_xor(ls, off, 32);
  if (lane == 0) s_red[wid] = ls;
  __syncthreads();
  float tot = s_red[0];
#pragma unroll
  for (int i = 1; i < 8; ++i) tot += s_red[i];
  const float inv = 1.0f / tot;

  float* __restrict__ arow = attn + (size_t)row * S_;
#pragma unroll
  for (int jj = 0; jj < 4; ++jj)
    arow[t + jj * 256] = p[jj] * inv;
}

// ---------------------------------------------------------------------------
// Kernel C: colsum[b,j] = sum_i attn[b,i,j]. Coalesced: consecutive threads
// read consecutive j within each i-row. grid = B*S/256 blocks of 256.
// ---------------------------------------------------------------------------
__global__ __launch_bounds__(256) void colsum_kernel(
    const float* __restrict__ attn, float* __restrict__ colsum) {
  const int g = blockIdx.x * 256 + threadIdx.x;   // 0 .. B*S-1
  const int b = g >> 10;
  const int j = g & (S_ - 1);
  const float* __restrict__ base = attn + (size_t)b * S_ * S_ + j;
  float s = 0.f;
  for (int i = 0; i < S_; ++i) s += base[(size_t)i * S_];
  colsum[g] = s;
}

// ---------------------------------------------------------------------------
// Kernel D: context[b,h] = sum_j colsum[b,j] * X[b,j,h]. grid = B*H/256.
// ---------------------------------------------------------------------------
__global__ __launch_bounds__(256) void context_kernel(
    const float* __restrict__ colsum, const float* __restrict__ X,
    float* __restrict__ ctx) {
  const int g = blockIdx.x * 256 + threadIdx.x;   // 0 .. B*H-1
  const int b = g >> 9;
  const int h = g & (H_ - 1);
  const float* __restrict__ xb = X + (size_t)b * S_ * H_;
  const float* __restrict__ cs = colsum + (size_t)b * S_;
  float s = 0.f;
  for (int j = 0; j < S_; ++j)
    s = fmaf(cs[j], xb[(size_t)j * H_ + h], s);
  ctx[g] = s;
}

extern "C" void kernel_launch(void* const* d_in, const int* in_sizes, int n_in,
                              void* d_out, int out_size, void* d_ws, size_t ws_size,
                              hipStream_t stream) {
  (void)in_sizes; (void)n_in; (void)out_size; (void)ws_size;
  const float* X = (const float*)d_in[0];   // (B,S,H)
  const float* W = (const float*)d_in[1];   // (H,A)
  const float* U = (const float*)d_in[2];   // (H,A)
  const float* V = (const float*)d_in[3];   // (A,)

  float* out  = (float*)d_out;
  float* ctx  = out;                         // (B,H)   = 1024 floats
  float* attn = out + (size_t)B_ * H_;       // (B,S,S) = 2M floats

  float* WX     = (float*)d_ws;                         // (B*S, A)
  float* UX     = WX + (size_t)B_ * S_ * A_;            // (B*S, A)
  float* colsum = UX + (size_t)B_ * S_ * A_;            // (B*S)

  // A: projections via f32 WMMA.  M-tiles=128 (grid.x*4 waves), N-tiles=8, z: W/U
  wmma_proj_kernel<<<dim3(32, 8, 2), 128, 0, stream>>>(X, W, U, WX, UX);
  // B: e + softmax, one block per (b,i) row
  attn_row_kernel<<<dim3(B_ * S_), 256, 0, stream>>>(WX, UX, V, attn);
  // C: column sums of attention
  colsum_kernel<<<dim3((B_ * S_) / 256), 256, 0, stream>>>(attn, colsum);
  // D: context matvec
  context_kernel<<<dim3((B_ * H_) / 256), 256, 0, stream>>>(colsum, X, ctx);
}